// TraditionalRAGRetriever_40707700031606
// MI455X (gfx1250) — compile-verified
//
#include <hip/hip_runtime.h>
#include <hip/hip_bf16.h>

typedef __attribute__((ext_vector_type(2))) float v2f;
typedef __attribute__((ext_vector_type(8))) float v8f;

#define DIM   768
#define SEQ   128
#define BATCH 32
#define TOPK  10
#define CHUNK 8192

// ---------------------------------------------------------------------------
// Kernel 1: mean-pool over seq, project (q @ W.T + b), L2-normalize -> qn[32,768]
// One block per batch row, 256 threads.
// ---------------------------------------------------------------------------
__global__ void rag_qproj(const float* __restrict__ query,
                          const float* __restrict__ W,
                          const float* __restrict__ bias,
                          float* __restrict__ qn) {
    const int b = blockIdx.x;
    const int t = threadIdx.x;
    __shared__ float qm[DIM];
    __shared__ float red[256];

    // mean pool over sequence
    for (int d = t; d < DIM; d += 256) {
        float s = 0.f;
        const float* p = query + (size_t)b * SEQ * DIM + d;
#pragma unroll 4
        for (int ss = 0; ss < SEQ; ++ss) s += p[(size_t)ss * DIM];
        qm[d] = s * (1.0f / (float)SEQ);
    }
    __syncthreads();

    // projection: proj[d] = dot(qm, W[d,:]) + bias[d]   (W.T multiply)
    float proj[3];
    float ssq = 0.f;
#pragma unroll
    for (int i = 0; i < 3; ++i) {
        const int d = t + i * 256;
        const float* wrow = W + (size_t)d * DIM;
        float acc = bias[d];
#pragma unroll 8
        for (int k = 0; k < DIM; ++k) acc = fmaf(qm[k], wrow[k], acc);
        proj[i] = acc;
        ssq += acc * acc;
    }
    red[t] = ssq;
    __syncthreads();
    for (int off = 128; off > 0; off >>= 1) {
        if (t < off) red[t] += red[t + off];
        __syncthreads();
    }
    const float inv = 1.0f / fmaxf(sqrtf(red[0]), 1e-12f);
#pragma unroll
    for (int i = 0; i < 3; ++i)
        qn[(size_t)b * DIM + t + i * 256] = proj[i] * inv;
}

// ---------------------------------------------------------------------------
// Kernel 2: similarity sim[32, nDocs] = qn @ normalize(docs).T via f32 WMMA.
// 8 waves/block, 16 docs per wave (one N tile), two M tiles (queries 0-15,16-31).
// Doc normalization folded in: accumulate sum-of-squares while streaming.
// ---------------------------------------------------------------------------
__global__ void rag_sim_wmma(const float* __restrict__ docs,
                             const float* __restrict__ qn,
                             float* __restrict__ sim,
                             int nDocs) {
    const int lane = threadIdx.x & 31;
    const int wave = threadIdx.x >> 5;
    const int nSub = lane & 15;   // N (doc) / M (query) index within tile
    const int half = lane >> 4;   // K-half select per ISA 16x4 f32 layout

    const int doc  = blockIdx.x * 128 + wave * 16 + nSub;
    const int docC = doc < nDocs ? doc : (nDocs - 1);   // clamp for loads

    const float* dptr = docs + (size_t)docC * DIM + 2 * half;
    const float* q0   = qn + (size_t)nSub * DIM + 2 * half;        // M-tile 0
    const float* q1   = qn + (size_t)(16 + nSub) * DIM + 2 * half; // M-tile 1

    v8f acc0 = {};
    v8f acc1 = {};
    float ssq = 0.f;

    for (int k = 0; k < DIM; k += 4) {
        v2f bv = *(const v2f*)(dptr + k);
        v2f a0 = *(const v2f*)(q0 + k);
        v2f a1 = *(const v2f*)(q1 + k);
        ssq = fmaf(bv.x, bv.x, ssq);
        ssq = fmaf(bv.y, bv.y, ssq);
        acc0 = __builtin_amdgcn_wmma_f32_16x16x4_f32(
            false, a0, false, bv, (short)0, acc0, false, false);
        acc1 = __builtin_amdgcn_wmma_f32_16x16x4_f32(
            false, a1, false, bv, (short)0, acc1, false, false);
    }

    // combine the two K-halves of the per-doc sum of squares
    const float ssT = ssq + __shfl_xor(ssq, 16, 32);
    const float inv = 1.0f / fmaxf(sqrtf(ssT), 1e-12f);

    if (doc < nDocs) {
#pragma unroll
        for (int j = 0; j < 8; ++j) {
            const int m = j + 8 * half;           // C/D layout: VGPR j, lane-half
            sim[(size_t)m * nDocs + doc]        = acc0[j] * inv;
            sim[(size_t)(m + 16) * nDocs + doc] = acc1[j] * inv;
        }
    }
}

// ---------------------------------------------------------------------------
// Top-k helpers: "greater" with jax tie-break (lower index wins on equal value)
// ---------------------------------------------------------------------------
__device__ __forceinline__ bool cand_gt(float av, int ai, float bv, int bi) {
    return (av > bv) || (av == bv && ai < bi);
}

__device__ __forceinline__ void cand_insert(float v, int id, float (&lv)[TOPK], int (&li)[TOPK]) {
    if (cand_gt(v, id, lv[TOPK - 1], li[TOPK - 1])) {
        lv[TOPK - 1] = v;
        li[TOPK - 1] = id;
        for (int j = TOPK - 1; j > 0 && cand_gt(lv[j], li[j], lv[j - 1], li[j - 1]); --j) {
            float tv = lv[j]; lv[j] = lv[j - 1]; lv[j - 1] = tv;
            int   ti = li[j]; li[j] = li[j - 1]; li[j - 1] = ti;
        }
    }
}

// 10-round block-wide merge of per-thread sorted lists. Writes winners to outV/outI.
__device__ __forceinline__ void block_merge_topk(float (&lv)[TOPK], int (&li)[TOPK],
                                                 float* outV, int* outI) {
    __shared__ float sv[256];
    __shared__ int   si[256];
    __shared__ int   st[256];
    const int t = threadIdx.x;
    int p = 0;
    for (int r = 0; r < TOPK; ++r) {
        sv[t] = (p < TOPK) ? lv[p] : -__builtin_inff();
        si[t] = (p < TOPK) ? li[p] : 0x7fffffff;
        st[t] = t;
        __syncthreads();
        for (int off = 128; off > 0; off >>= 1) {
            if (t < off) {
                if (cand_gt(sv[t + off], si[t + off], sv[t], si[t])) {
                    sv[t] = sv[t + off];
                    si[t] = si[t + off];
                    st[t] = st[t + off];
                }
            }
            __syncthreads();
        }
        if (t == 0) {
            outV[r] = sv[0];
            outI[r] = si[0];
        }
        const int winner = st[0];
        __syncthreads();   // everyone has read sv/si/st[0] before next round
        if (t == winner) p++;
    }
}

// ---------------------------------------------------------------------------
// Kernel 3: per-(row, chunk) top-10 candidates.  grid = (nChunks, 32)
// ---------------------------------------------------------------------------
__global__ void rag_topk_stage1(const float* __restrict__ sim, int nDocs,
                                float* __restrict__ cVal, int* __restrict__ cIdx,
                                int nChunks) {
    const int row   = blockIdx.y;
    const int chunk = blockIdx.x;
    const int t     = threadIdx.x;
    const float* s  = sim + (size_t)row * nDocs;

    float lv[TOPK];
    int   li[TOPK];
#pragma unroll
    for (int j = 0; j < TOPK; ++j) { lv[j] = -__builtin_inff(); li[j] = 0x7fffffff; }

    const int base = chunk * CHUNK;
    for (int i = 0; i < CHUNK / 256; ++i) {
        const int e = base + t + i * 256;
        float v = (e < nDocs) ? s[e] : -__builtin_inff();
        int  id = (e < nDocs) ? e : 0x7fffffff;
        cand_insert(v, id, lv, li);
    }

    float* outV = cVal + ((size_t)row * nChunks + chunk) * TOPK;
    int*   outI = cIdx + ((size_t)row * nChunks + chunk) * TOPK;
    block_merge_topk(lv, li, outV, outI);
}

// ---------------------------------------------------------------------------
// Kernel 4: reduce chunk candidates to final top-10 indices per row.
// ---------------------------------------------------------------------------
__global__ void rag_topk_stage2(const float* __restrict__ cVal,
                                const int* __restrict__ cIdx,
                                int nChunks, int* __restrict__ outIdx) {
    const int row = blockIdx.x;
    const int t   = threadIdx.x;
    const int M   = nChunks * TOPK;

    float lv[TOPK];
    int   li[TOPK];
#pragma unroll
    for (int j = 0; j < TOPK; ++j) { lv[j] = -__builtin_inff(); li[j] = 0x7fffffff; }

    for (int e = t; e < M; e += 256) {
        float v = cVal[(size_t)row * M + e];
        int  id = cIdx[(size_t)row * M + e];
        cand_insert(v, id, lv, li);
    }

    __shared__ float fv[TOPK];
    __shared__ int   fi[TOPK];
    block_merge_topk(lv, li, fv, fi);
    __syncthreads();
    if (t < TOPK) outIdx[row * TOPK + t] = fi[t];
}

// ---------------------------------------------------------------------------
extern "C" void kernel_launch(void* const* d_in, const int* in_sizes, int n_in,
                              void* d_out, int out_size, void* d_ws, size_t ws_size,
                              hipStream_t stream) {
    const float* query = (const float*)d_in[0];   // [32,128,768]
    const float* docs  = (const float*)d_in[1];   // [N,768] raw (normalized on the fly)
    const float* W     = (const float*)d_in[2];   // [768,768]
    const float* bias  = (const float*)d_in[3];   // [768]

    const int nDocs = in_sizes[1] / DIM;

    // d_out = [indices (32*10 ints, bit-cast) | sims (32*nDocs f32)]
    int*   outIdx = (int*)d_out;
    float* sim    = (float*)d_out + BATCH * TOPK;

    // workspace: qn [32,768] | cVal [32*nChunks*10] | cIdx [32*nChunks*10]
    const int nChunks = (nDocs + CHUNK - 1) / CHUNK;
    float* qn   = (float*)d_ws;
    float* cVal = qn + BATCH * DIM;
    int*   cIdx = (int*)(cVal + (size_t)BATCH * nChunks * TOPK);

    rag_qproj<<<BATCH, 256, 0, stream>>>(query, W, bias, qn);

    const int simBlocks = (nDocs + 127) / 128;
    rag_sim_wmma<<<simBlocks, 256, 0, stream>>>(docs, qn, sim, nDocs);

    dim3 g1(nChunks, BATCH);
    rag_topk_stage1<<<g1, 256, 0, stream>>>(sim, nDocs, cVal, cIdx, nChunks);
    rag_topk_stage2<<<BATCH, 256, 0, stream>>>(cVal, cIdx, nChunks, outIdx);
}